// HyperGsysUinGINConv_27831388078177
// MI455X (gfx1250) — compile-verified
//
#include <hip/hip_runtime.h>

#define NN 100000   // nodes
#define NE 50000    // edges
#define NNZ_ 800000
#define D 256

#define PAIR_STRIDE 160   // dwords per k-pair row in LDS; 160 % 64 == 32 -> the two
                          // EXEC halves (pair p vs p+1) hit disjoint bank halves

typedef __attribute__((ext_vector_type(2))) float v2f;
typedef __attribute__((ext_vector_type(8))) float v8f;

// ---------------------------------------------------------------------------
// GEMM: Xp[M, 256] = X[M, 256] @ W[256, 256] in f32 via V_WMMA_F32_16X16X4_F32
// Block: 256 threads = 8 waves. Tile: 128 rows x 64 cols.
// W column slab staged in LDS k-pair-interleaved:
//   ldsW[p*PAIR_STRIDE + n*2 + 0] = W[2p  ][n0+n]
//   ldsW[p*PAIR_STRIDE + n*2 + 1] = W[2p+1][n0+n]
// so each B fragment {W[k][n], W[k+1][n]} is one aligned ds_load_b64.
// ---------------------------------------------------------------------------
__global__ __launch_bounds__(256) void gemm_xp(const float* __restrict__ X,
                                               const float* __restrict__ W,
                                               float* __restrict__ Xp)
{
    __shared__ float ldsW[(D / 2) * PAIR_STRIDE];   // 128 * 160 * 4B = 80 KB

    const int n0 = blockIdx.y * 64;

    // Stage W[:, n0:n0+64] interleaved. 128 pairs x 16 col-quads = 2048 tasks,
    // 256 threads -> 8 tasks each. Each task: two float4 global loads, two
    // float4 LDS stores (interleaved).
    #pragma unroll
    for (int i = 0; i < 8; ++i) {
        int t  = i * 256 + threadIdx.x;   // 0..2047
        int p  = t >> 4;                  // k-pair index 0..127
        int c4 = (t & 15) << 2;           // col quad base 0..60
        float4 w0 = *(const float4*)&W[(size_t)(2 * p)     * D + n0 + c4];
        float4 w1 = *(const float4*)&W[(size_t)(2 * p + 1) * D + n0 + c4];
        float* dst = &ldsW[p * PAIR_STRIDE + c4 * 2];
        *(float4*)(dst + 0) = make_float4(w0.x, w1.x, w0.y, w1.y);
        *(float4*)(dst + 4) = make_float4(w0.z, w1.z, w0.w, w1.w);
    }
    __syncthreads();

    const int wave = threadIdx.x >> 5;
    const int lane = threadIdx.x & 31;
    const int l16  = lane & 15;
    const int half = lane >> 4;            // 0: lanes 0-15, 1: lanes 16-31
    const int klo  = half * 2;             // A/B fragment K phase per ISA layout

    const int m0 = blockIdx.x * 128 + wave * 16;
    int mA = m0 + l16;
    if (mA > NN - 1) mA = NN - 1;          // clamp: keep EXEC all-1s for WMMA

    v8f acc0 = {}, acc1 = {}, acc2 = {}, acc3 = {};

    const float* xrow = X + (size_t)mA * D;

    #pragma unroll 4
    for (int k0 = 0; k0 < D; k0 += 4) {
        // A fragment 16x4: lane holds X[m][k0+klo], X[m][k0+klo+1] (8B load)
        v2f a = *(const v2f*)(xrow + k0 + klo);

        // B fragments: one ds_load_b64 per 16-col subtile, pair-adjacent data
        const float* wp = &ldsW[((k0 + klo) >> 1) * PAIR_STRIDE + l16 * 2];
        v2f b0 = *(const v2f*)(wp +  0);
        v2f b1 = *(const v2f*)(wp + 32);
        v2f b2 = *(const v2f*)(wp + 64);
        v2f b3 = *(const v2f*)(wp + 96);

        acc0 = __builtin_amdgcn_wmma_f32_16x16x4_f32(false, a, false, b0, (short)0, acc0, false, false);
        acc1 = __builtin_amdgcn_wmma_f32_16x16x4_f32(false, a, false, b1, (short)0, acc1, false, false);
        acc2 = __builtin_amdgcn_wmma_f32_16x16x4_f32(false, a, false, b2, (short)0, acc2, false, false);
        acc3 = __builtin_amdgcn_wmma_f32_16x16x4_f32(false, a, false, b3, (short)0, acc3, false, false);
    }

    // C/D layout: VGPR r holds row m0 + r + 8*half, col = n0 + 16*j + l16
    #pragma unroll
    for (int r = 0; r < 8; ++r) {
        int m = m0 + r + 8 * half;
        if (m < NN) {
            float* out = Xp + (size_t)m * D + n0 + l16;
            out[ 0] = acc0[r];
            out[16] = acc1[r];
            out[32] = acc2[r];
            out[48] = acc3[r];
        }
    }
}

// ---------------------------------------------------------------------------
// Zero Xe/Xv accumulators in workspace (float4 stores).
// ---------------------------------------------------------------------------
__global__ __launch_bounds__(256) void zero_ws(float4* __restrict__ p, int n4)
{
    int i = blockIdx.x * 256 + threadIdx.x;
    if (i < n4) p[i] = make_float4(0.f, 0.f, 0.f, 0.f);
}

// ---------------------------------------------------------------------------
// Pass 1: Xe_raw[e] += Xp[v]  (degE folded into pass 2)
// One block per nnz; thread d handles channel d. Xp/Xe are L2-resident.
// ---------------------------------------------------------------------------
__global__ __launch_bounds__(256) void scatter_n2e(const float* __restrict__ Xp,
                                                   const int* __restrict__ node_idx,
                                                   const int* __restrict__ edge_idx,
                                                   float* __restrict__ Xe)
{
    int n = blockIdx.x;
    int v = node_idx[n];
    int e = edge_idx[n];
    int d = threadIdx.x;
    atomicAdd(&Xe[(size_t)e * D + d], Xp[(size_t)v * D + d]);
}

// ---------------------------------------------------------------------------
// Pass 2: Xv[v] += Xe_raw[e] * degE[e]
// ---------------------------------------------------------------------------
__global__ __launch_bounds__(256) void scatter_e2v(const float* __restrict__ Xe,
                                                   const float* __restrict__ degE,
                                                   const int* __restrict__ node_idx,
                                                   const int* __restrict__ edge_idx,
                                                   float* __restrict__ Xv)
{
    int n = blockIdx.x;
    int v = node_idx[n];
    int e = edge_idx[n];
    int d = threadIdx.x;
    float s = degE[e];
    atomicAdd(&Xv[(size_t)v * D + d], Xe[(size_t)e * D + d] * s);
}

// ---------------------------------------------------------------------------
// out = (1 + eps) * Xp + Xv * degV   (Xp already resident in d_out)
// ---------------------------------------------------------------------------
__global__ __launch_bounds__(256) void finalize(float* __restrict__ out,
                                                const float* __restrict__ Xv,
                                                const float* __restrict__ degV,
                                                const float* __restrict__ eps)
{
    size_t i = (size_t)blockIdx.x * 256 + threadIdx.x;   // grid = NN blocks
    float g = 1.0f + eps[0];
    int v = (int)(i >> 8);
    out[i] = g * out[i] + Xv[i] * degV[v];
}

// ---------------------------------------------------------------------------
// Launch
// inputs: 0:X [NN*256] 1:W [256*256] 2:eps [1] 3:degE [NE] 4:degV [NN]
//         5:node_idx [NNZ] 6:edge_idx [NNZ]
// workspace layout: Xe [NE*256] floats | Xv [NN*256] floats  (153.6 MB)
// ---------------------------------------------------------------------------
extern "C" void kernel_launch(void* const* d_in, const int* in_sizes, int n_in,
                              void* d_out, int out_size, void* d_ws, size_t ws_size,
                              hipStream_t stream)
{
    const float* X        = (const float*)d_in[0];
    const float* W        = (const float*)d_in[1];
    const float* eps      = (const float*)d_in[2];
    const float* degE     = (const float*)d_in[3];
    const float* degV     = (const float*)d_in[4];
    const int*   node_idx = (const int*)d_in[5];
    const int*   edge_idx = (const int*)d_in[6];

    float* Xp = (float*)d_out;                    // Xp lives in d_out
    float* Xe = (float*)d_ws;                     // NE * 256 floats
    float* Xv = Xe + (size_t)NE * D;              // NN * 256 floats

    // 1) GEMM: Xp = X @ W
    dim3 g1((NN + 127) / 128, D / 64);
    gemm_xp<<<g1, 256, 0, stream>>>(X, W, Xp);

    // 2) zero scatter accumulators
    int n4 = (NE + NN) * (D / 4);                 // float4 count
    zero_ws<<<(n4 + 255) / 256, 256, 0, stream>>>((float4*)d_ws, n4);

    // 3) nodes -> hyperedges
    scatter_n2e<<<NNZ_, 256, 0, stream>>>(Xp, node_idx, edge_idx, Xe);

    // 4) hyperedges -> nodes (degE folded in)
    scatter_e2v<<<NNZ_, 256, 0, stream>>>(Xe, degE, node_idx, edge_idx, Xv);

    // 5) residual combine
    finalize<<<NN, 256, 0, stream>>>(Xp, Xv, degV, eps);
}